// SpherePool_31731218383015
// MI455X (gfx1250) — compile-verified
//
#include <hip/hip_runtime.h>

// SpherePool: out[bt, j, c] = max_{k<7} x[bt, index[j,k], c]
// B*T=16, N=40962, N_OUT=10242, C=256, K=7, all fp32, index int64.
//
// Pure bandwidth-bound gather-max. Mapping: one wave32 per (bt, j) output
// row; each lane handles 8 channels as two float4 chunks -> every row read
// is two coalesced global_load_b128 per wave. j varies fastest so each
// 41MB bt-slice of the tensor gets L2 (192MB) reuse; output uses
// non-temporal stores to avoid polluting L2.

typedef float v4f __attribute__((ext_vector_type(4)));

constexpr int BT    = 16;      // B*T
constexpr int NIN   = 40962;
constexpr int NOUT  = 10242;
constexpr int C     = 256;
constexpr int K     = 7;
constexpr int WAVES_PER_BLOCK = 8;          // 256 threads, wave32
constexpr int PAIRS = BT * NOUT;            // 163872 output rows
constexpr int C4    = C / 4;                // 64 float4 chunks per row

__global__ __launch_bounds__(256) void sphere_pool_max_kernel(
    const float* __restrict__ x,
    const long long* __restrict__ idx,
    float* __restrict__ out)
{
    const unsigned tid  = threadIdx.x;
    const unsigned wave = tid >> 5;          // uniform per wave32
    const unsigned lane = tid & 31;

    const unsigned pair = blockIdx.x * WAVES_PER_BLOCK + wave;
    if (pair >= (unsigned)PAIRS) return;

    // j fastest across consecutive waves/blocks -> bt-slice locality in L2
    const unsigned bt = pair / NOUT;
    const unsigned j  = pair - bt * NOUT;

    const long long* __restrict__ jrow = idx + (size_t)j * K;

    // Prefetch the index stream ~64 rows ahead (gfx1250 global_prefetch_b8).
    __builtin_prefetch(jrow + (size_t)64 * K, 0, 1);

    // Gather the 7 row indices (uniform address per wave; coalesces in L0).
    long long r[K];
#pragma unroll
    for (int k = 0; k < K; ++k) r[k] = jrow[k];

    const v4f* __restrict__ base4 =
        (const v4f*)(x + (size_t)bt * ((size_t)NIN * C));

    // Each lane covers float4 chunks {lane, lane+32} of the 64-chunk row.
    // Fully unrolled: 14 independent b128 loads in flight before the maxes.
    v4f m0 = base4[(size_t)r[0] * C4 + lane];
    v4f m1 = base4[(size_t)r[0] * C4 + lane + 32];
#pragma unroll
    for (int k = 1; k < K; ++k) {
        const v4f a = base4[(size_t)r[k] * C4 + lane];
        const v4f b = base4[(size_t)r[k] * C4 + lane + 32];
        m0.x = fmaxf(m0.x, a.x); m0.y = fmaxf(m0.y, a.y);
        m0.z = fmaxf(m0.z, a.z); m0.w = fmaxf(m0.w, a.w);
        m1.x = fmaxf(m1.x, b.x); m1.y = fmaxf(m1.y, b.y);
        m1.z = fmaxf(m1.z, b.z); m1.w = fmaxf(m1.w, b.w);
    }

    // Streaming (write-once) output: non-temporal b128 stores keep L2 free
    // for the reusable tensor slice.
    v4f* o = (v4f*)(out + (size_t)pair * C);
    __builtin_nontemporal_store(m0, o + lane);
    __builtin_nontemporal_store(m1, o + lane + 32);
}

extern "C" void kernel_launch(void* const* d_in, const int* in_sizes, int n_in,
                              void* d_out, int out_size, void* d_ws, size_t ws_size,
                              hipStream_t stream) {
    (void)in_sizes; (void)n_in; (void)out_size; (void)d_ws; (void)ws_size;
    const float*     x   = (const float*)d_in[0];
    const long long* idx = (const long long*)d_in[1];   // reference index is int64
    float*           out = (float*)d_out;

    const int blocks = (PAIRS + WAVES_PER_BLOCK - 1) / WAVES_PER_BLOCK;  // 20484
    sphere_pool_max_kernel<<<blocks, 256, 0, stream>>>(x, idx, out);
}